// CodebookEMA3D_64381559767328
// MI455X (gfx1250) — compile-verified
//
#include <hip/hip_runtime.h>
#include <hip/hip_bf16.h>

// Problem constants (from reference)
#define KC    1024          // num codebook vectors
#define DD    256           // latent dim
#define NV    32768         // number of latent vectors (32*32*32)
#define HWSZ  1024          // H*W
#define DECAY 0.99f
#define BETA  0.25f
#define EPSV  1e-5f

typedef __attribute__((ext_vector_type(2))) float v2f;
typedef __attribute__((ext_vector_type(8))) float v8f;

// ---------------------------------------------------------------------------
// Kernel 0: zero scratch accumulators (dw, counts, nsum) and the loss slot.
// ---------------------------------------------------------------------------
__global__ void vq_init_kernel(float* __restrict__ dw, float* __restrict__ counts,
                               float* __restrict__ nsum, float* __restrict__ loss) {
    int i = blockIdx.x * 256 + threadIdx.x;
    if (i < KC * DD) dw[i] = 0.0f;
    if (i < KC)      counts[i] = 0.0f;
    if (i == 0) { *nsum = 0.0f; *loss = 0.0f; }
}

// ---------------------------------------------------------------------------
// Kernel 1: enorm2[k] = sum_d embedding[k,d]^2  (one wave per code row)
// ---------------------------------------------------------------------------
__global__ void vq_enorm_kernel(const float* __restrict__ emb, float* __restrict__ enorm2) {
    int wave = (blockIdx.x * blockDim.x + threadIdx.x) >> 5;   // 0..KC-1
    int lane = threadIdx.x & 31;
    const float* row = emb + (size_t)wave * DD;
    float s = 0.0f;
#pragma unroll
    for (int j = 0; j < DD / 32; ++j) { float v = row[lane + 32 * j]; s += v * v; }
#pragma unroll
    for (int m = 16; m >= 1; m >>= 1) s += __shfl_xor(s, m, 32);
    if (lane == 0) enorm2[wave] = s;
}

// ---------------------------------------------------------------------------
// Kernel 2: WMMA fp32 GEMM + fused argmax of (z.e - 0.5*|e|^2).
// One wave owns TWO 16-row tiles of zf (M=32) so each B fragment load from
// the embedding feeds two WMMAs (register-level B reuse -> half the L2
// traffic per FLOP). One wave per 32-thread block; grid = NV/32.
// ---------------------------------------------------------------------------
#define LDS_STRIDE 260   // 256 + 4 floats pad -> conflict-free stage & ds_load_b64

__global__ __launch_bounds__(32) void vq_argmin_wmma_kernel(
    const float* __restrict__ z, const float* __restrict__ emb,
    const float* __restrict__ enorm2,
    int* __restrict__ idx_i32, float* __restrict__ idx_f32) {

    __shared__ float ldsA[2][16 * LDS_STRIDE];   // 33,280 B

    const int lane = threadIdx.x;        // 0..31
    const int m    = lane & 15;          // row within 16-tile / col within 16-tile
    const int dl   = lane >> 4;          // half-select (K-pair for WMMA frags)

    const int rowBase = blockIdx.x * 32;                 // never straddles a batch image
    const int b   = rowBase >> 10;
    const int hw0 = rowBase & 1023;
    const float* zb = z + (size_t)b * (DD * HWSZ) + hw0; // + d*HWSZ + (t*16 + m)

    // Stage two A tiles [16 x 256] to LDS: lane L loads (m=L%16, d=d0+L/16):
    // fixed d -> 16 consecutive hw addresses (coalesced 64B per d).
#pragma unroll
    for (int t = 0; t < 2; ++t) {
        for (int d0 = 0; d0 < DD; d0 += 2) {
            int d = d0 + dl;
            ldsA[t][m * LDS_STRIDE + d] = zb[(size_t)d * HWSZ + t * 16 + m];
        }
    }
    __syncthreads();   // single-wave WG: lowers to S_NOP, acts as compiler fence

    float bestS0[8], bestS1[8];
    int   bestC0[8], bestC1[8];
#pragma unroll
    for (int v = 0; v < 8; ++v) {
        bestS0[v] = -3.4e38f; bestC0[v] = 0;
        bestS1[v] = -3.4e38f; bestC1[v] = 0;
    }

    for (int ct = 0; ct < KC / 16; ++ct) {
        const int col = ct * 16 + m;
        const float* brow = emb + (size_t)col * DD + 2 * dl;   // B frag base (embedding^T)
        v8f acc0 = {};
        v8f acc1 = {};
#pragma unroll 8
        for (int kk = 0; kk < DD / 4; ++kk) {
            // B 4x16 f32 frag, loaded ONCE, feeds both M-tiles
            v2f bf = *(const v2f*)(brow + kk * 4);
            // A 16x4 f32 frags: lanes 0-15 K={0,1}, lanes 16-31 K={2,3}
            v2f a0 = *(const v2f*)&ldsA[0][m * LDS_STRIDE + kk * 4 + 2 * dl];
            v2f a1 = *(const v2f*)&ldsA[1][m * LDS_STRIDE + kk * 4 + 2 * dl];
            acc0 = __builtin_amdgcn_wmma_f32_16x16x4_f32(
                false, a0, false, bf, (short)0, acc0, false, false);
            acc1 = __builtin_amdgcn_wmma_f32_16x16x4_f32(
                false, a1, false, bf, (short)0, acc1, false, false);
        }
        const float halfE = 0.5f * enorm2[col];
#pragma unroll
        for (int v = 0; v < 8; ++v) {
            float s0 = acc0[v] - halfE;               // lane holds (row = v + 8*dl, col)
            float s1 = acc1[v] - halfE;
            if (s0 > bestS0[v]) { bestS0[v] = s0; bestC0[v] = col; }
            if (s1 > bestS1[v]) { bestS1[v] = s1; bestC1[v] = col; }
        }
    }

    // Reduce across the 16 lanes that share a row (masks < 16 keep halves apart).
#pragma unroll
    for (int t = 0; t < 2; ++t) {
#pragma unroll
        for (int v = 0; v < 8; ++v) {
            float s = t ? bestS1[v] : bestS0[v];
            int   c = t ? bestC1[v] : bestC0[v];
#pragma unroll
            for (int msk = 1; msk < 16; msk <<= 1) {
                float so = __shfl_xor(s, msk, 32);
                int   co = __shfl_xor(c, msk, 32);
                if (so > s || (so == s && co < c)) { s = so; c = co; }  // tie -> lowest idx
            }
            if (m == 0) {
                int row = rowBase + t * 16 + dl * 8 + v;  // lane0 rows 0-7, lane16 rows 8-15
                idx_i32[row] = c;
                idx_f32[row] = (float)c;
            }
        }
    }
}

// ---------------------------------------------------------------------------
// Kernel 3: gather z_q (write back in [B,D,H,W]), loss, dw scatter, counts.
// block = 256 threads as (32 n-lanes x 8 d-slices); grid = NV/32.
// ---------------------------------------------------------------------------
__global__ __launch_bounds__(256) void vq_gather_kernel(
    const float* __restrict__ z, const float* __restrict__ emb,
    const int* __restrict__ idx,
    float* __restrict__ zq_out, float* __restrict__ loss,
    float* __restrict__ dw, float* __restrict__ counts) {

    const int tx = threadIdx.x & 31;
    const int ty = threadIdx.x >> 5;
    const int n  = blockIdx.x * 32 + tx;
    const int b  = n >> 10, hw = n & 1023;
    const int code = idx[n];
    if (ty == 0) atomicAdd(&counts[code], 1.0f);

    const float* erow = emb + (size_t)code * DD;
    float lsum = 0.0f;
    for (int d = ty; d < DD; d += 8) {
        float e = erow[d];
        size_t gi = (size_t)b * (DD * HWSZ) + (size_t)d * HWSZ + hw;
        float zv = z[gi];
        zq_out[gi] = e;                          // coalesced along hw
        float df = e - zv;
        lsum += df * df;
        atomicAdd(&dw[(size_t)code * DD + d], zv);
    }

    __shared__ float red[256];
    red[threadIdx.x] = lsum;
    __syncthreads();
    for (int s = 128; s > 0; s >>= 1) {
        if (threadIdx.x < s) red[threadIdx.x] += red[threadIdx.x + s];
        __syncthreads();
    }
    if (threadIdx.x == 0)
        atomicAdd(loss, red[0] * (BETA / (float)(NV * DD)));
}

// ---------------------------------------------------------------------------
// Kernel 4: cs_raw = ema_cs*decay + (1-decay)*counts ; nsum = sum(cs_raw)
// ---------------------------------------------------------------------------
__global__ __launch_bounds__(256) void vq_cs_kernel(
    const float* __restrict__ ecs, const float* __restrict__ counts,
    float* __restrict__ cs_raw, float* __restrict__ nsum) {
    int k = blockIdx.x * 256 + threadIdx.x;
    float v = ecs[k] * DECAY + (1.0f - DECAY) * counts[k];
    cs_raw[k] = v;
    __shared__ float red[256];
    red[threadIdx.x] = v;
    __syncthreads();
    for (int s = 128; s > 0; s >>= 1) {
        if (threadIdx.x < s) red[threadIdx.x] += red[threadIdx.x + s];
        __syncthreads();
    }
    if (threadIdx.x == 0) atomicAdd(nsum, red[0]);
}

// ---------------------------------------------------------------------------
// Kernel 5: finalize new_cs, new_ema_w, new_embedding. grid = KC, block = DD.
// ---------------------------------------------------------------------------
__global__ __launch_bounds__(256) void vq_final_kernel(
    const float* __restrict__ cs_raw, const float* __restrict__ nsum,
    const float* __restrict__ emaw, const float* __restrict__ dw,
    float* __restrict__ new_emb, float* __restrict__ new_cs,
    float* __restrict__ new_emaw) {
    int k = blockIdx.x;
    int d = threadIdx.x;
    float nS = *nsum;
    float cs = (cs_raw[k] + EPSV) / (nS + (float)KC * EPSV) * nS;
    if (d == 0) new_cs[k] = cs;
    size_t i = (size_t)k * DD + d;
    float wv = emaw[i] * DECAY + (1.0f - DECAY) * dw[i];
    new_emaw[i] = wv;
    new_emb[i]  = wv / cs;
}

// ---------------------------------------------------------------------------
extern "C" void kernel_launch(void* const* d_in, const int* in_sizes, int n_in,
                              void* d_out, int out_size, void* d_ws, size_t ws_size,
                              hipStream_t stream) {
    const float* z    = (const float*)d_in[0];   // [32,256,32,32]
    const float* emb  = (const float*)d_in[1];   // [1024,256]
    const float* ecs  = (const float*)d_in[2];   // [1024]
    const float* emaw = (const float*)d_in[3];   // [1024,256]

    float* out = (float*)d_out;
    // Output regions (floats), concatenated in return order:
    float* o_zq    = out;                               // 8388608
    float* o_idx   = out + 8388608;                     // 32768
    float* o_loss  = out + 8388608 + 32768;             // 1
    float* o_nemb  = o_loss + 1;                        // 262144
    float* o_ncs   = o_nemb + 262144;                   // 1024
    float* o_nemaw = o_ncs + 1024;                      // 262144

    // Workspace layout (bytes)
    char* ws = (char*)d_ws;
    int*   w_idx    = (int*)(ws);                       // 32768 * 4 = 131072
    float* w_enorm  = (float*)(ws + 131072);            // 1024 * 4
    float* w_counts = (float*)(ws + 135168);            // 1024 * 4
    float* w_csraw  = (float*)(ws + 139264);            // 1024 * 4
    float* w_nsum   = (float*)(ws + 143360);            // 4 (padded to 256)
    float* w_dw     = (float*)(ws + 143616);            // 262144 * 4 = 1 MB

    // 0) zero accumulators
    vq_init_kernel<<<1024, 256, 0, stream>>>(w_dw, w_counts, w_nsum, o_loss);
    // 1) per-code squared norms
    vq_enorm_kernel<<<128, 256, 0, stream>>>(emb, w_enorm);
    // 2) fp32 WMMA score GEMM + fused argmin (M=32 per wave, B reuse)
    vq_argmin_wmma_kernel<<<NV / 32, 32, 0, stream>>>(z, emb, w_enorm, w_idx, o_idx);
    // 3) gather / z_q / loss / EMA scatter stats
    vq_gather_kernel<<<NV / 32, 256, 0, stream>>>(z, emb, w_idx, o_zq, o_loss,
                                                  w_dw, w_counts);
    // 4) cluster-size EMA + global sum
    vq_cs_kernel<<<KC / 256, 256, 0, stream>>>(ecs, w_counts, w_csraw, w_nsum);
    // 5) finalize embedding / cs / ema_w
    vq_final_kernel<<<KC, 256, 0, stream>>>(w_csraw, w_nsum, emaw, w_dw,
                                            o_nemb, o_ncs, o_nemaw);
}